// GraphAttentionLayer_25074019074120
// MI455X (gfx1250) — compile-verified
//
#include <hip/hip_runtime.h>
#include <hip/hip_bf16.h>
#include <math.h>

// ---------------------------------------------------------------------------
// GAT layer for MI455X (gfx1250, wave32).
// Sizes fixed by the reference: B=2, N=4096, Fin=Fout=64, k=20.
// Strategy: never materialize the N x N distance / attention matrices.
//   k1: Wh = x @ W via V_WMMA_F32_16X16X4_F32 (exact f32 matrix path)
//   k2: per-node sq = |pos|^2, f1 = Wh . a[:64], f2 = Wh . a[64:]
//   k3: kNN top-21 per row, register-resident sorted lists + wave merge
//   k4: 20-neighbor softmax + sparse aggregation + pos-MLP + ELU
// ---------------------------------------------------------------------------

typedef __attribute__((ext_vector_type(2))) float v2f;
typedef __attribute__((ext_vector_type(8))) float v8f;

#define NB       2
#define NN       4096
#define BN       (NB * NN)     // 8192 nodes total
#define FDIM     64
#define KNN      20
#define LSZ      (KNN + 1)     // top-21 (self + 20 neighbors)
#define LRELU_A  0.2f
#define WT_PITCH 66            // padded LDS pitch: bank = (2n + k) % 64, no conflicts

// -------------------------------------------------------------------- k1 ---
// Wh[8192 x 64] = x[8192 x 64] @ W[64 x 64], f32 WMMA (16x16 tiles, K=4 steps).
// One wave per 16-row tile; 8 waves / block.
// W is staged TRANSPOSED in LDS (Wt[n][k], pitch 66) so each lane's B-frag
// (K=kk, kk+1 at fixed N) is one contiguous, 8-byte-aligned ds_load_b64
// straight into an even VGPR pair — no repack moves, half the DS traffic.
// A frag (16x4 f32, 2 VGPRs): lane l, vgpr v -> (M = l%16, K = 2*(l>>4)+v)
// B frag (4x16 f32, 2 VGPRs): lane l, vgpr v -> (K = 2*(l>>4)+v, N = l%16)
// D (16x16 f32, 8 VGPRs):     lane l, vgpr v -> (M = v + 8*(l>>4), N = l%16)
__global__ void k_wh(const float* __restrict__ x, const float* __restrict__ W,
                     float* __restrict__ Wh) {
  __shared__ __align__(16) float Wt[FDIM * WT_PITCH];
  const int tid = threadIdx.x;
  for (int i = tid; i < FDIM * FDIM; i += 256) {
    const int k = i >> 6, n = i & 63;          // W is (k, n) row-major
    Wt[n * WT_PITCH + k] = W[i];               // store transposed
  }
  __syncthreads();

  const int wave = tid >> 5;
  const int lane = tid & 31;
  const int lm = lane & 15;
  const int hi = lane >> 4;
  const int r0 = (blockIdx.x * 8 + wave) * 16; // 64 blocks * 8 waves * 16 rows

  v8f acc[4];
  #pragma unroll
  for (int t = 0; t < 4; ++t) acc[t] = (v8f){0.f,0.f,0.f,0.f,0.f,0.f,0.f,0.f};

  #pragma unroll
  for (int kb = 0; kb < FDIM; kb += 4) {
    const int kk = kb + 2 * hi;                // this lane's K base (even)
    v2f afrag = *(const v2f*)(x + (size_t)(r0 + lm) * FDIM + kk);
    #pragma unroll
    for (int t = 0; t < 4; ++t) {
      v2f bfrag = *(const v2f*)&Wt[(t * 16 + lm) * WT_PITCH + kk];
      acc[t] = __builtin_amdgcn_wmma_f32_16x16x4_f32(
          false, afrag, false, bfrag, (short)0, acc[t], false, false);
    }
  }

  #pragma unroll
  for (int t = 0; t < 4; ++t)
    #pragma unroll
    for (int v = 0; v < 8; ++v)
      Wh[(size_t)(r0 + v + 8 * hi) * FDIM + t * 16 + lm] = acc[t][v];
}

// -------------------------------------------------------------------- k2 ---
// Per global node n: sq[n] = |pos[n]|^2; f1[n] = Wh[n].a[0:64]; f2 = .a[64:128]
__global__ void k_node(const float* __restrict__ pos, const float* __restrict__ Wh,
                       const float* __restrict__ a, float* __restrict__ sq,
                       float* __restrict__ f1, float* __restrict__ f2) {
  const int n = blockIdx.x * blockDim.x + threadIdx.x;   // 0..8191
  if (n >= BN) return;
  const float px = pos[n * 3 + 0], py = pos[n * 3 + 1], pz = pos[n * 3 + 2];
  sq[n] = px * px + py * py + pz * pz;
  const float* w = Wh + (size_t)n * FDIM;
  float s1 = 0.f, s2 = 0.f;
  #pragma unroll 8
  for (int c = 0; c < FDIM; ++c) {
    const float v = w[c];
    s1 += v * a[c];
    s2 += v * a[FDIM + c];
  }
  f1[n] = s1;
  f2[n] = s2;
}

// -------------------------------------------------------------------- k3 ---
// One wave32 per row. Each lane scans 4096/32 = 128 columns maintaining a
// sorted (ascending d2, tie: ascending idx) top-21 in registers (static
// indexing only -> SROA to VGPRs). Then 21 rounds of cross-wave head-merge:
// round 0 = self (dropped, matching idx[:, :, 1:]), rounds 1..20 -> neighbors.
__device__ __forceinline__ unsigned fkey(float f) {
  unsigned u = __float_as_uint(f);
  return u ^ ((u >> 31) ? 0xFFFFFFFFu : 0x80000000u);   // order-preserving
}

__global__ void k_knn(const float* __restrict__ pos, const float* __restrict__ sq,
                      int* __restrict__ knn) {
  const int wave = threadIdx.x >> 5;
  const int lane = threadIdx.x & 31;
  const int row = blockIdx.x * 8 + wave;                 // 0..8191
  const int b = row >> 12;                               // row / 4096
  const int n = row & (NN - 1);
  const float* pb = pos + (size_t)b * NN * 3;
  const float* sqb = sq + b * NN;
  const float px = pb[n * 3 + 0], py = pb[n * 3 + 1], pz = pb[n * 3 + 2];
  const float sqn = sqb[n];

  float ld[LSZ];
  int   li[LSZ];
  #pragma unroll
  for (int i = 0; i < LSZ; ++i) { ld[i] = INFINITY; li[i] = 0x7FFFFFFF; }

  for (int m = lane; m < NN; m += 32) {
    const float qx = pb[m * 3 + 0], qy = pb[m * 3 + 1], qz = pb[m * 3 + 2];
    const float dot = px * qx + py * qy + pz * qz;
    const float d2 = sqn + sqb[m] - 2.0f * dot;          // reference formula
    if (d2 <= ld[LSZ - 1]) {                             // reject fast path
      float cd = d2; int ci = m;
      #pragma unroll
      for (int i = 0; i < LSZ; ++i) {
        const bool less = (cd < ld[i]) || (cd == ld[i] && ci < li[i]);
        const float td = ld[i]; const int ti = li[i];
        if (less) { ld[i] = cd; li[i] = ci; cd = td; ci = ti; }
      }
    }
  }

  // 21-round merge of 32 sorted lists: pick global min head each round.
  #pragma unroll
  for (int r = 0; r < LSZ; ++r) {
    unsigned bk = fkey(ld[0]);
    unsigned bi = (unsigned)li[0];
    int bl = lane;
    #pragma unroll
    for (int off = 16; off > 0; off >>= 1) {
      const unsigned ok = (unsigned)__shfl_xor((int)bk, off, 32);
      const unsigned oi = (unsigned)__shfl_xor((int)bi, off, 32);
      const int ol = __shfl_xor(bl, off, 32);
      if (ok < bk || (ok == bk && (oi < bi || (oi == bi && ol < bl)))) {
        bk = ok; bi = oi; bl = ol;
      }
    }
    if (r > 0 && lane == 0) knn[row * KNN + (r - 1)] = (int)bi;  // skip self
    if (lane == bl) {                                            // pop head
      #pragma unroll
      for (int i = 0; i < LSZ - 1; ++i) { ld[i] = ld[i + 1]; li[i] = li[i + 1]; }
      ld[LSZ - 1] = INFINITY; li[LSZ - 1] = 0x7FFFFFFF;
    }
  }
}

// -------------------------------------------------------------------- k4 ---
// One wave32 per node. Lanes 0..19 own one neighbor each (score + softmax),
// every lane owns 2 of the 64 output channels (c, c+32) for the gather-sum.
__global__ void k_aggr(const float* __restrict__ Wh, const float* __restrict__ pos,
                       const float* __restrict__ Wp, const float* __restrict__ bp,
                       const float* __restrict__ f1, const float* __restrict__ f2,
                       const int* __restrict__ knn, float* __restrict__ out) {
  const int wave = threadIdx.x >> 5;
  const int lane = threadIdx.x & 31;
  const int node = blockIdx.x * 8 + wave;                // 0..8191
  const int b = node >> 12;

  int m = 0;
  float e = -INFINITY;
  if (lane < KNN) {
    m = knn[node * KNN + lane];                          // within-batch index
    const float ev = f1[node] + f2[b * NN + m];
    e = (ev > 0.f) ? ev : LRELU_A * ev;                  // leaky relu pre-mask
  }
  // softmax over the 20 live entries (masked NEG_BIG entries underflow to 0)
  float mx = e;
  #pragma unroll
  for (int off = 16; off > 0; off >>= 1) mx = fmaxf(mx, __shfl_xor(mx, off, 32));
  float w = (lane < KNN) ? expf(e - mx) : 0.f;
  float s = w;
  #pragma unroll
  for (int off = 16; off > 0; off >>= 1) s += __shfl_xor(s, off, 32);
  const float attn = w / s;

  const int c0 = lane, c1 = lane + 32;
  float acc0 = 0.f, acc1 = 0.f;
  #pragma unroll
  for (int j = 0; j < KNN; ++j) {
    const float aj = __shfl(attn, j, 32);
    const int mj = __shfl(m, j, 32);
    const float* wr = Wh + (size_t)(b * NN + mj) * FDIM;
    acc0 += aj * wr[c0];
    acc1 += aj * wr[c1];
  }

  // relu(pos @ Wp + bp); Wp is (3, 64) row-major
  const float px = pos[node * 3 + 0], py = pos[node * 3 + 1], pz = pos[node * 3 + 2];
  float p0 = px * Wp[c0] + py * Wp[FDIM + c0] + pz * Wp[2 * FDIM + c0] + bp[c0];
  float p1 = px * Wp[c1] + py * Wp[FDIM + c1] + pz * Wp[2 * FDIM + c1] + bp[c1];
  p0 = fmaxf(p0, 0.f);
  p1 = fmaxf(p1, 0.f);

  const float h0 = acc0 + p0, h1 = acc1 + p1;
  out[(size_t)node * FDIM + c0] = (h0 > 0.f) ? h0 : expm1f(h0);   // elu
  out[(size_t)node * FDIM + c1] = (h1 > 0.f) ? h1 : expm1f(h1);
}

// ---------------------------------------------------------------------------
extern "C" void kernel_launch(void* const* d_in, const int* in_sizes, int n_in,
                              void* d_out, int out_size, void* d_ws, size_t ws_size,
                              hipStream_t stream) {
  const float* x   = (const float*)d_in[0];   // (2, 4096, 64)
  const float* pos = (const float*)d_in[1];   // (2, 4096, 3)
  const float* W   = (const float*)d_in[2];   // (64, 64)
  const float* a   = (const float*)d_in[3];   // (128, 1)
  const float* Wp  = (const float*)d_in[4];   // (3, 64)
  const float* bp  = (const float*)d_in[5];   // (64,)
  float* out = (float*)d_out;                 // (2, 4096, 64)

  // workspace layout (~2.9 MB)
  float* Wh = (float*)d_ws;                   // 8192 * 64
  float* sq = Wh + (size_t)BN * FDIM;         // 8192
  float* f1 = sq + BN;                        // 8192
  float* f2 = f1 + BN;                        // 8192
  int* knn  = (int*)(f2 + BN);                // 8192 * 20

  k_wh  <<<BN / (16 * 8), 256, 0, stream>>>(x, W, Wh);
  k_node<<<BN / 256,      256, 0, stream>>>(pos, Wh, a, sq, f1, f2);
  k_knn <<<BN / 8,        256, 0, stream>>>(pos, sq, knn);
  k_aggr<<<BN / 8,        256, 0, stream>>>(Wh, pos, Wp, bp, f1, f2, knn, out);
}